// GATv2Conv_graphgym_8169027797782
// MI455X (gfx1250) — compile-verified
//
#include <hip/hip_runtime.h>
#include <math.h>

#define HEADS     4
#define C_OUT     64
#define D_IN      128
#define HC        256   // HEADS * C_OUT
#define NEG_SLOPE 0.2f

typedef float v2f __attribute__((ext_vector_type(2)));
typedef float v8f __attribute__((ext_vector_type(8)));

// ---- monotonic int encoding for float atomicMax ----
__device__ __forceinline__ int enc_f32(float f) {
  int i = __float_as_int(f);
  return i >= 0 ? i : (i ^ 0x7fffffff);
}
__device__ __forceinline__ float dec_f32(int i) {
  return __int_as_float(i >= 0 ? i : (i ^ 0x7fffffff));
}

// ---------------------------------------------------------------------------
// Kernel 1: init out = bias, nodeMax = enc(-3e38), denom = 0
// ---------------------------------------------------------------------------
__global__ void gat_init(float* __restrict__ out, const float* __restrict__ bias,
                         int* __restrict__ nodeMax, float* __restrict__ denom,
                         int n_nodes) {
  int t = blockIdx.x * blockDim.x + threadIdx.x;
  if (t < n_nodes * C_OUT) out[t] = bias[t & (C_OUT - 1)];
  if (t < n_nodes * HEADS) {
    nodeMax[t] = enc_f32(-3.0e38f);
    denom[t]   = 0.0f;
  }
}

// ---------------------------------------------------------------------------
// Kernel 2: xl = x @ Wl, xr = x @ Wr  via V_WMMA_F32_16X16X4_F32
// One wave per 16x16 output tile. 32 col-tiles (16 for Wl, 16 for Wr).
// A 16x4 f32: lane l -> row = l&15, VGPR0/1 = K = (l>>4)*2 (+1)
// B 4x16 f32: lane l -> col = l&15, VGPR0/1 = K = (l>>4)*2 (+1)
// D 16x16 f32: VGPR r, lane l -> row (l>>4)*8 + r, col l&15
// ---------------------------------------------------------------------------
__global__ void __launch_bounds__(128)
gat_gemm_wmma(const float* __restrict__ x,
              const float* __restrict__ Wl,
              const float* __restrict__ Wr,
              float* __restrict__ xl,
              float* __restrict__ xr,
              int n_nodes, int n_row_tiles) {
  const int lane      = threadIdx.x & 31;
  const int wave      = blockIdx.x * 4 + (threadIdx.x >> 5);
  const int rowTile   = wave >> 5;          // 32 col-tiles per row-tile
  if (rowTile >= n_row_tiles) return;       // wave-uniform, EXEC stays full
  const int ct        = wave & 31;
  const float* __restrict__ W = (ct & 16) ? Wr : Wl;
  float*       __restrict__ Y = (ct & 16) ? xr : xl;
  const int colBase   = (ct & 15) * 16;

  const int half = lane >> 4;               // selects K pair within step-of-4
  const int l15  = lane & 15;
  int arow = rowTile * 16 + l15;
  if (arow >= n_nodes) arow = n_nodes - 1;  // tail safety (N%16==0 here)
  const int bcol = colBase + l15;

  const float* __restrict__ ap = x + (size_t)arow * D_IN + half * 2;
  const float* __restrict__ bp = W + (size_t)(half * 2) * HC + bcol;

  v8f c = {};
#pragma unroll 8
  for (int k0 = 0; k0 < D_IN; k0 += 4) {
    v2f a = *(const v2f*)ap;                // 8B aligned (even K offset)
    v2f b;
    b.x = bp[0];
    b.y = bp[HC];
    c = __builtin_amdgcn_wmma_f32_16x16x4_f32(false, a, false, b,
                                              (short)0, c, false, false);
    ap += 4;
    bp += 4 * HC;
  }

  const int orow = rowTile * 16 + half * 8;
  float* __restrict__ yp = Y + (size_t)orow * HC + bcol;
  if (rowTile * 16 + 16 <= n_nodes) {
    // full tile (wave-uniform): 8 plain stores, immediate row offsets
#pragma unroll
    for (int r = 0; r < 8; ++r) yp[(size_t)r * HC] = c[r];
  } else {
    // partial tail tile: per-row guard
#pragma unroll
    for (int r = 0; r < 8; ++r)
      if (orow + r < n_nodes) yp[(size_t)r * HC] = c[r];
  }
}

// ---------------------------------------------------------------------------
// Kernel 3: edge scores e[e,h] = sum_c lrelu(xl[s,h,c]+xr[d,h,c]) * att[h,c]
// One wave (32 lanes) per edge; lane covers 8 contiguous channels ->
// head h = lane>>3; butterfly reduce within 8-lane groups.
// ---------------------------------------------------------------------------
__global__ void __launch_bounds__(256)
gat_edge_scores(const int* __restrict__ esrc, const int* __restrict__ edst,
                const float* __restrict__ xl, const float* __restrict__ xr,
                const float* __restrict__ att,
                float* __restrict__ escore, int* __restrict__ nodeMax,
                int n_edges, int n_nodes) {
  const int lane = threadIdx.x & 31;
  const int e    = (blockIdx.x * blockDim.x + threadIdx.x) >> 5;
  const int etot = n_edges + n_nodes;
  if (e >= etot) return;                    // wave-uniform
  int s, d;
  if (e < n_edges) { s = esrc[e]; d = edst[e]; }
  else             { s = d = e - n_edges; }

  const float* __restrict__ pl = xl + (size_t)s * HC + lane * 8;
  const float* __restrict__ pr = xr + (size_t)d * HC + lane * 8;
  const float* __restrict__ pa = att + lane * 8;

  float4 l0 = *(const float4*)(pl);
  float4 l1 = *(const float4*)(pl + 4);
  float4 r0 = *(const float4*)(pr);
  float4 r1 = *(const float4*)(pr + 4);
  float4 a0 = *(const float4*)(pa);
  float4 a1 = *(const float4*)(pa + 4);

  float partial = 0.0f;
  {
    float m;
    m = l0.x + r0.x; partial += (m > 0.f ? m : NEG_SLOPE * m) * a0.x;
    m = l0.y + r0.y; partial += (m > 0.f ? m : NEG_SLOPE * m) * a0.y;
    m = l0.z + r0.z; partial += (m > 0.f ? m : NEG_SLOPE * m) * a0.z;
    m = l0.w + r0.w; partial += (m > 0.f ? m : NEG_SLOPE * m) * a0.w;
    m = l1.x + r1.x; partial += (m > 0.f ? m : NEG_SLOPE * m) * a1.x;
    m = l1.y + r1.y; partial += (m > 0.f ? m : NEG_SLOPE * m) * a1.y;
    m = l1.z + r1.z; partial += (m > 0.f ? m : NEG_SLOPE * m) * a1.z;
    m = l1.w + r1.w; partial += (m > 0.f ? m : NEG_SLOPE * m) * a1.w;
  }
  // reduce within each 8-lane head group
  partial += __shfl_xor(partial, 1);
  partial += __shfl_xor(partial, 2);
  partial += __shfl_xor(partial, 4);

  if ((lane & 7) == 0) {
    const int h = lane >> 3;
    escore[(size_t)e * HEADS + h] = partial;
    atomicMax(&nodeMax[(size_t)d * HEADS + h], enc_f32(partial));
  }
}

// ---------------------------------------------------------------------------
// Kernel 4: expw = exp(e - max[dst,h]); denom[dst,h] += expw   (in-place)
// ---------------------------------------------------------------------------
__global__ void __launch_bounds__(256)
gat_edge_exp(const int* __restrict__ edst,
             float* __restrict__ escore, const int* __restrict__ nodeMax,
             float* __restrict__ denom, int n_edges, int n_nodes) {
  const int etot = n_edges + n_nodes;
  int t = blockIdx.x * blockDim.x + threadIdx.x;
  if (t >= etot * HEADS) return;
  const int e = t >> 2;
  const int h = t & (HEADS - 1);
  const int d = (e < n_edges) ? edst[e] : (e - n_edges);
  const float m = dec_f32(nodeMax[(size_t)d * HEADS + h]);
  const float w = __expf(escore[t] - m);
  escore[t] = w;
  atomicAdd(&denom[(size_t)d * HEADS + h], w);
}

// ---------------------------------------------------------------------------
// Kernel 5: out[d,c] += (1/H) * sum_h alpha[e,h] * xl[s,h,c]
// One wave per edge; lane handles channels c=lane and c=lane+32.
// ---------------------------------------------------------------------------
__global__ void __launch_bounds__(256)
gat_scatter(const int* __restrict__ esrc, const int* __restrict__ edst,
            const float* __restrict__ xl, const float* __restrict__ expw,
            const float* __restrict__ denom, float* __restrict__ out,
            int n_edges, int n_nodes) {
  const int lane = threadIdx.x & 31;
  const int e    = (blockIdx.x * blockDim.x + threadIdx.x) >> 5;
  const int etot = n_edges + n_nodes;
  if (e >= etot) return;
  int s, d;
  if (e < n_edges) { s = esrc[e]; d = edst[e]; }
  else             { s = d = e - n_edges; }

  float4 w = *(const float4*)(expw + (size_t)e * HEADS);   // 16B aligned
  float alpha[HEADS];
  alpha[0] = w.x / denom[(size_t)d * HEADS + 0];
  alpha[1] = w.y / denom[(size_t)d * HEADS + 1];
  alpha[2] = w.z / denom[(size_t)d * HEADS + 2];
  alpha[3] = w.w / denom[(size_t)d * HEADS + 3];

  const float* __restrict__ pl = xl + (size_t)s * HC;
  float acc0 = 0.f, acc1 = 0.f;
#pragma unroll
  for (int h = 0; h < HEADS; ++h) {
    acc0 += alpha[h] * pl[h * C_OUT + lane];
    acc1 += alpha[h] * pl[h * C_OUT + 32 + lane];
  }
  atomicAdd(&out[(size_t)d * C_OUT + lane],      0.25f * acc0);
  atomicAdd(&out[(size_t)d * C_OUT + 32 + lane], 0.25f * acc1);
}

// ---------------------------------------------------------------------------
extern "C" void kernel_launch(void* const* d_in, const int* in_sizes, int n_in,
                              void* d_out, int out_size, void* d_ws, size_t ws_size,
                              hipStream_t stream) {
  const float* x    = (const float*)d_in[0];
  const int*   ei   = (const int*)d_in[1];     // int edge_index [2,E]
  const float* Wl   = (const float*)d_in[2];
  const float* Wr   = (const float*)d_in[3];
  const float* att  = (const float*)d_in[4];
  const float* bias = (const float*)d_in[5];

  const int N = in_sizes[0] / D_IN;
  const int E = in_sizes[1] / 2;
  const int* esrc = ei;
  const int* edst = ei + E;
  float* out = (float*)d_out;

  // workspace layout
  char*  ws  = (char*)d_ws;
  size_t off = 0;
  float* xl      = (float*)(ws + off); off += (size_t)N * HC * sizeof(float);
  float* xr      = (float*)(ws + off); off += (size_t)N * HC * sizeof(float);
  int*   nodeMax = (int*)  (ws + off); off += (size_t)N * HEADS * sizeof(int);
  float* denom   = (float*)(ws + off); off += (size_t)N * HEADS * sizeof(float);
  float* escore  = (float*)(ws + off); off += (size_t)(E + N) * HEADS * sizeof(float);
  (void)ws_size; (void)n_in; (void)out_size;

  const int etot = E + N;

  // 1) init
  {
    int total = N * C_OUT;
    gat_init<<<(total + 255) / 256, 256, 0, stream>>>(out, bias, nodeMax, denom, N);
  }
  // 2) GEMM (WMMA f32)
  {
    int rowTiles = (N + 15) / 16;
    int waves    = rowTiles * 32;            // 32 col-tiles per row-tile
    gat_gemm_wmma<<<(waves + 3) / 4, 128, 0, stream>>>(x, Wl, Wr, xl, xr, N, rowTiles);
  }
  // 3) edge scores + segment max
  gat_edge_scores<<<(etot + 7) / 8, 256, 0, stream>>>(esrc, edst, xl, xr, att,
                                                      escore, nodeMax, E, N);
  // 4) exp + segment sum
  gat_edge_exp<<<(etot * HEADS + 255) / 256, 256, 0, stream>>>(edst, escore, nodeMax,
                                                               denom, E, N);
  // 5) normalized weighted scatter (head-mean folded in)
  gat_scatter<<<(etot + 7) / 8, 256, 0, stream>>>(esrc, edst, xl, escore, denom,
                                                  out, E, N);
}